// DSSnetwork_28123445854597
// MI455X (gfx1250) — compile-verified
//
#include <hip/hip_runtime.h>
#include <hip/hip_bf16.h>

// ---------------------------------------------------------------------------
// DSS-GNN forward for MI455X (gfx1250). fp32 end-to-end; GEMMs use
// v_wmma_f32_16x16x4_f32 with fragment-ready LDS weight layout (single
// ds_load_b64 per B fragment, bank-conflict-free). Edge aggregation via
// hardware f32 atomics (both 64MB buffers are L2-resident on 192MB L2).
// ---------------------------------------------------------------------------

#define N_NODES 512
#define N_SUB   512
#define NT      (N_NODES * N_SUB)      // 262144
#define E_SUB   2097152
#define E_ORIG  8192
#define EMB     64
#define LAYERS  4
#define TASKS   10

typedef __attribute__((ext_vector_type(2))) float v2f;
typedef __attribute__((ext_vector_type(8))) float v8f;

__device__ __forceinline__ v8f wmma4(v2f a, v2f b, v8f c) {
    // D = A(16x4 f32) * B(4x16 f32) + C(16x16 f32)
    return __builtin_amdgcn_wmma_f32_16x16x4_f32(
        /*neg_a=*/false, a, /*neg_b=*/false, b,
        /*c_mod=*/(short)0, c, /*reuse_a=*/false, /*reuse_b=*/false);
}

__device__ __forceinline__ void atomAddF(float* p, float v) {
    unsafeAtomicAdd(p, v);   // hardware global_atomic_add_f32
}

// --------------------------------------------------------------------------
// agg[dst] += h[src] over edges; one thread = (edge, 4-column chunk)
// --------------------------------------------------------------------------
__global__ void edge_scatter(const float* __restrict__ h, float* __restrict__ agg,
                             const int* __restrict__ src, const int* __restrict__ dst,
                             int nEdges) {
    long idx = (long)blockIdx.x * blockDim.x + threadIdx.x;
    long total = (long)nEdges * 16;
    if (idx >= total) return;
    int e  = (int)(idx >> 4);
    int ch = (int)(idx & 15);
    long so  = (long)src[e] * EMB + ch * 4;
    long dof = (long)dst[e] * EMB + ch * 4;
    float4 v = *reinterpret_cast<const float4*>(h + so);
    atomAddF(agg + dof + 0, v.x);
    atomAddF(agg + dof + 1, v.y);
    atomAddF(agg + dof + 2, v.z);
    atomAddF(agg + dof + 3, v.w);
}

// --------------------------------------------------------------------------
// P[r,:] = A1[r,:] @ W1^T + A2[r,:] @ W2^T + bias      (all [*,64] x [64,64])
// One wave produces a 16x64 strip via v_wmma_f32_16x16x4_f32.
// LDS layout (per matrix): pair-row p = k/2, pitch 160 floats:
//   lds[p*160 + col*2 + (k&1)] = B[k][col] = W[col][k]
// -> B fragment {B[kk][col], B[kk+1][col]} is ONE contiguous ds_load_b64,
//    and the two half-waves (pair-rows p, p+1) hit disjoint 32-bank halves.
// P may alias A1 (each wave reads/writes only its own 16 rows).
// --------------------------------------------------------------------------
#define BPITCH 160

__global__ void __launch_bounds__(256) gemm_dual64(
    const float* __restrict__ A1, const float* __restrict__ A2,
    const float* __restrict__ W1, const float* __restrict__ W2,
    const float* __restrict__ bias, float* __restrict__ P, int rows)
{
    __shared__ float lds1[32 * BPITCH];
    __shared__ float lds2[32 * BPITCH];
    for (int i = threadIdx.x; i < 4096; i += 256) {
        int e = i >> 6, k = i & 63;
        int a = (k >> 1) * BPITCH + e * 2 + (k & 1);
        lds1[a] = W1[i];   // B1[k][e]
        lds2[a] = W2[i];   // B2[k][e]
    }
    __syncthreads();

    int wave = blockIdx.x * (blockDim.x >> 5) + (threadIdx.x >> 5);
    int lane = threadIdx.x & 31;
    int r0 = wave * 16;
    if (r0 >= rows) return;

    int mrow = lane & 15;            // A row within tile (both half-waves)
    int koff = (lane >> 4) << 1;     // K sub-offset: 0 (lanes 0-15) / 2 (16-31)
    int col0 = lane & 15;            // B/C column within 16-wide tile

    v8f acc[4];
    acc[0] = {}; acc[1] = {}; acc[2] = {}; acc[3] = {};

    const float* a1p = A1 + (long)(r0 + mrow) * EMB + koff;
    const float* a2p = A2 + (long)(r0 + mrow) * EMB + koff;

#pragma unroll
    for (int k0 = 0; k0 < 64; k0 += 4) {
        v2f a1 = *reinterpret_cast<const v2f*>(a1p + k0);
        v2f a2 = *reinterpret_cast<const v2f*>(a2p + k0);
        int base = ((k0 + koff) >> 1) * BPITCH + col0 * 2;
#pragma unroll
        for (int nt = 0; nt < 4; ++nt) {
            v2f b1 = *reinterpret_cast<const v2f*>(&lds1[base + nt * 32]);
            acc[nt] = wmma4(a1, b1, acc[nt]);
            v2f b2 = *reinterpret_cast<const v2f*>(&lds2[base + nt * 32]);
            acc[nt] = wmma4(a2, b2, acc[nt]);
        }
    }

    int rshift = (lane >> 4) << 3;   // C rows: +0 (lanes 0-15) / +8 (16-31)
#pragma unroll
    for (int nt = 0; nt < 4; ++nt) {
        int col = nt * 16 + col0;
        float bb = bias[col];
#pragma unroll
        for (int i = 0; i < 8; ++i) {
            P[(long)(r0 + rshift + i) * EMB + col] = acc[nt][i] + bb;
        }
    }
}

// --------------------------------------------------------------------------
// Column sum and sum-of-squares: S[0..63]=sum, S[64..127]=sumsq
// --------------------------------------------------------------------------
__global__ void colstats(const float* __restrict__ P, float* __restrict__ S, int rows) {
    int c = threadIdx.x & 63, g = threadIdx.x >> 6;
    float s1 = 0.f, s2 = 0.f;
    for (long r = (long)blockIdx.x * 4 + g; r < rows; r += (long)gridDim.x * 4) {
        float v = P[r * EMB + c];
        s1 += v; s2 += v * v;
    }
    __shared__ float sm[512];
    sm[threadIdx.x] = s1;
    sm[256 + threadIdx.x] = s2;
    __syncthreads();
    if (g == 0) {
        float a = sm[c] + sm[c + 64] + sm[c + 128] + sm[c + 192];
        float b = sm[256 + c] + sm[256 + c + 64] + sm[256 + c + 128] + sm[256 + c + 192];
        atomAddF(S + c, a);
        atomAddF(S + 64 + c, b);
    }
}

// BN fold: scale = gamma*rsqrt(var+eps); shift = beta - mu*scale
__global__ void bn_coeffs(const float* __restrict__ S,
                          const float* __restrict__ gamma, const float* __restrict__ beta,
                          float invn, float* __restrict__ AB) {
    int c = threadIdx.x;
    float mu  = S[c] * invn;
    float var = S[64 + c] * invn - mu * mu;
    float sc  = gamma[c] * rsqrtf(var + 1e-5f);
    AB[c]      = sc;
    AB[64 + c] = beta[c] - mu * sc;
}

// h[t,c] = relu( bn1(h1[t,c]) + bn2(h2[t % 512, c]) )   (in-place over h)
__global__ void fuse_update(const float* __restrict__ h1, const float* __restrict__ h2,
                            const float* __restrict__ AB1, const float* __restrict__ AB2,
                            float* __restrict__ h, long total) {
    long idx = (long)blockIdx.x * blockDim.x + threadIdx.x;
    if (idx >= total) return;
    int c = (int)(idx & 63);
    long t = idx >> 6;
    int j = (int)(t & (N_NODES - 1));
    float a = h1[idx] * AB1[c] + AB1[64 + c];
    float b = h2[j * EMB + c] * AB2[c] + AB2[64 + c];
    float v = a + b;
    h[idx] = v > 0.f ? v : 0.f;
}

// x_sum[j,c] = (1/512) * sum_s h[s*512 + j, c]   (one block per j)
__global__ void subgraph_mean(const float* __restrict__ h, float* __restrict__ xsum) {
    int j = blockIdx.x;
    int c = threadIdx.x & 63, g = threadIdx.x >> 6;
    float acc = 0.f;
    for (int s = g; s < N_SUB; s += 4)
        acc += h[((long)s * N_NODES + j) * EMB + c];
    __shared__ float sm[256];
    sm[threadIdx.x] = acc;
    __syncthreads();
    if (g == 0)
        xsum[j * EMB + c] = (sm[c] + sm[c + 64] + sm[c + 128] + sm[c + 192]) * (1.0f / N_SUB);
}

// per-row log_softmax over 64 columns (512 blocks x 64 threads)
__global__ void logsoftmax64(const float* __restrict__ x, float* __restrict__ z) {
    int r = blockIdx.x, c = threadIdx.x;
    __shared__ float sm[64];
    float v = x[r * EMB + c];
    sm[c] = v; __syncthreads();
#pragma unroll
    for (int off = 32; off > 0; off >>= 1) {
        if (c < off) sm[c] = fmaxf(sm[c], sm[c + off]);
        __syncthreads();
    }
    float m = sm[0]; __syncthreads();
    sm[c] = expf(v - m); __syncthreads();
#pragma unroll
    for (int off = 32; off > 0; off >>= 1) {
        if (c < off) sm[c] += sm[c + off];
        __syncthreads();
    }
    z[r * EMB + c] = v - m - logf(sm[0]);
}

// out[r,:] = relu(z[r,:] @ W1^T + b1) @ W2^T + b2   (512 blocks x 128 threads)
__global__ void mlp_head(const float* __restrict__ z,
                         const float* __restrict__ W1, const float* __restrict__ b1,
                         const float* __restrict__ W2, const float* __restrict__ b2,
                         float* __restrict__ out) {
    int r = blockIdx.x, c = threadIdx.x;
    __shared__ float t[2 * EMB];
    float acc = b1[c];
#pragma unroll 8
    for (int k = 0; k < EMB; ++k) acc += z[r * EMB + k] * W1[c * EMB + k];
    t[c] = acc > 0.f ? acc : 0.f;
    __syncthreads();
    if (c < TASKS) {
        float o = b2[c];
#pragma unroll 8
        for (int k = 0; k < 2 * EMB; ++k) o += t[k] * W2[c * 2 * EMB + k];
        out[r * TASKS + c] = o;
    }
}

// ---------------------------------------------------------------------------

extern "C" void kernel_launch(void* const* d_in, const int* in_sizes, int n_in,
                              void* d_out, int out_size, void* d_ws, size_t ws_size,
                              hipStream_t stream) {
    const float* x       = (const float*)d_in[0];
    const float* Wrel    = (const float*)d_in[1];
    const float* brel    = (const float*)d_in[2];
    const float* Wroot   = (const float*)d_in[3];
    const float* bng     = (const float*)d_in[4];
    const float* bnb     = (const float*)d_in[5];
    const float* Wrel_s  = (const float*)d_in[6];
    const float* brel_s  = (const float*)d_in[7];
    const float* Wroot_s = (const float*)d_in[8];
    const float* bnsg    = (const float*)d_in[9];
    const float* bnsb    = (const float*)d_in[10];
    const float* W1      = (const float*)d_in[11];
    const float* b1      = (const float*)d_in[12];
    const float* W2      = (const float*)d_in[13];
    const float* b2      = (const float*)d_in[14];
    const int*   ei      = (const int*)d_in[15];   // [2, E_SUB]
    const int*   oei     = (const int*)d_in[16];   // [2, E_ORIG]
    float* out = (float*)d_out;

    const int* e_src = ei;
    const int* e_dst = ei + E_SUB;
    const int* o_src = oei;
    const int* o_dst = oei + E_ORIG;

    float* ws   = (float*)d_ws;
    float* h    = ws;                       // NT*64 floats
    float* agg  = h + (long)NT * EMB;       // NT*64 floats (also h1pre)
    float* xsum = agg + (long)NT * EMB;     // 512*64
    float* aggs = xsum + N_NODES * EMB;     // 512*64 (also h2pre)
    float* zbuf = aggs + N_NODES * EMB;     // 512*64
    float* Sa   = zbuf + N_NODES * EMB;     // 128
    float* Sb   = Sa + 128;                 // 128
    float* AB1  = Sb + 128;                 // 128
    float* AB2  = AB1 + 128;                // 128

    // h = x (do not mutate inputs)
    hipMemcpyAsync(h, x, (size_t)NT * EMB * sizeof(float),
                   hipMemcpyDeviceToDevice, stream);

    const long totalElems = (long)NT * EMB;
    const int edgeBlocksMain = (int)(((long)E_SUB * 16 + 255) / 256);
    const int edgeBlocksOrig = (int)(((long)E_ORIG * 16 + 255) / 256);
    const int gemmBlocksMain = (NT / 16) / 8;     // 8 waves/block, 16 rows/wave
    const int gemmBlocksSub  = (N_NODES / 16) / 8;

    for (int l = 0; l < LAYERS; ++l) {
        const float* Wr  = Wrel    + (long)l * EMB * EMB;
        const float* Wt  = Wroot   + (long)l * EMB * EMB;
        const float* br  = brel    + (long)l * EMB;
        const float* g1  = bng     + (long)l * EMB;
        const float* bb1 = bnb     + (long)l * EMB;
        const float* Wrs = Wrel_s  + (long)l * EMB * EMB;
        const float* Wts = Wroot_s + (long)l * EMB * EMB;
        const float* brs = brel_s  + (long)l * EMB;
        const float* g2  = bnsg    + (long)l * EMB;
        const float* bb2 = bnsb    + (long)l * EMB;

        // main branch: agg = segment_sum(h[src] -> dst)
        hipMemsetAsync(agg, 0, (size_t)NT * EMB * sizeof(float), stream);
        edge_scatter<<<edgeBlocksMain, 256, 0, stream>>>(h, agg, e_src, e_dst, E_SUB);

        // subgraph mean of current h (independent of agg)
        subgraph_mean<<<N_NODES, 256, 0, stream>>>(h, xsum);

        // h1pre = agg @ Wr^T + h @ Wt^T + br   (written over agg)
        gemm_dual64<<<gemmBlocksMain, 256, 0, stream>>>(agg, h, Wr, Wt, br, agg, NT);

        hipMemsetAsync(Sa, 0, 128 * sizeof(float), stream);
        colstats<<<1024, 256, 0, stream>>>(agg, Sa, NT);
        bn_coeffs<<<1, 64, 0, stream>>>(Sa, g1, bb1, 1.0f / (float)NT, AB1);

        // subgraph branch: aggs = segment_sum(xsum[osrc] -> odst)
        hipMemsetAsync(aggs, 0, (size_t)N_NODES * EMB * sizeof(float), stream);
        edge_scatter<<<edgeBlocksOrig, 256, 0, stream>>>(xsum, aggs, o_src, o_dst, E_ORIG);

        // h2pre = aggs @ Wrs^T + xsum @ Wts^T + brs   (written over aggs)
        gemm_dual64<<<gemmBlocksSub, 256, 0, stream>>>(aggs, xsum, Wrs, Wts, brs, aggs, N_NODES);

        hipMemsetAsync(Sb, 0, 128 * sizeof(float), stream);
        colstats<<<128, 256, 0, stream>>>(aggs, Sb, N_NODES);
        bn_coeffs<<<1, 64, 0, stream>>>(Sb, g2, bb2, 1.0f / (float)N_NODES, AB2);

        // h = relu(bn1(h1pre) + bn2(h2pre[t % 512]))  in-place
        fuse_update<<<(int)((totalElems + 255) / 256), 256, 0, stream>>>(
            agg, aggs, AB1, AB2, h, totalElems);
    }

    // head: x_nodes = smean(h); z = log_softmax(x_nodes); out = relu(z@W1^T+b1)@W2^T+b2
    subgraph_mean<<<N_NODES, 256, 0, stream>>>(h, xsum);
    logsoftmax64<<<N_NODES, 64, 0, stream>>>(xsum, zbuf);
    mlp_head<<<N_NODES, 128, 0, stream>>>(zbuf, W1, b1, W2, b2, out);
}